// WorldClassMetabolicEncoder_10995116278230
// MI455X (gfx1250) — compile-verified
//
#include <hip/hip_runtime.h>
#include <hip/hip_bf16.h>

// ---------------------------------------------------------------------------
// GAT encoder (VGAE) for MI455X / gfx1250.
//   - GEMMs use V_WMMA_F32_16X16X4_F32 (fp32, matches reference numerics).
//     A-tile (16 rows, shared by all 8 waves of the block) is staged in LDS
//     once -> 8x reduction in A-side global traffic; WMMA A-fragments come
//     from conflict-free ds_load_b64 (row pitch K+4 => bank stride 4).
//   - Edge phase (HBM/L2-bound hot path, ~8-9 GB over 4 layers): 3-pass
//     dst-softmax with atomicMax(ordered-uint) + atomicAdd(f32), then
//     128-wide weighted scatter-add per edge. xh (51 MB) is L2-resident
//     (192 MB L2), so gather/scatter atomics mostly stay on-chip.
//   - LayerNorm fused with residual+bias: one wave32 per node, shfl reduce.
// ---------------------------------------------------------------------------

#define NN      100000
#define EE      1600000
#define ETOT    (EE + NN)      // edges + self loops = 1,700,000
#define F_IN    256
#define DD      128
#define HH      8
#define CC      16
#define LL      4
#define NEG_SLOPE 0.2f
#define LN_EPS  1e-5f

typedef __attribute__((ext_vector_type(2))) float v2f;
typedef __attribute__((ext_vector_type(8))) float v8f;

// ------------------------- GEMM: Y[N,128] = X[N,K] @ W[128,K]^T (+bias) -----
// Block = 256 threads = 8 waves. Each wave owns one 16x16 output tile:
// rows = blockIdx.x*16 .. +15, cols = wave*16 .. +15. K stepped by 4 via
// V_WMMA_F32_16X16X4_F32. Fragment layouts per CDNA5 ISA 7.12.2.
template <int K>
__global__ __launch_bounds__(256) void gemm_wmma_k(
    const float* __restrict__ X, const float* __restrict__ W,
    const float* __restrict__ bias, float* __restrict__ Y) {
  constexpr int KP = K + 4;            // LDS row pitch: stride 4 banks mod 64
  __shared__ float As[16 * KP];        // 16 x K A tile (<= 16.6 KB)

  const int wave = threadIdx.x >> 5;
  const int lane = threadIdx.x & 31;
  const int row0 = blockIdx.x * 16;
  const int col0 = wave * 16;
  const int m    = lane & 15;           // A row / B col within tile
  const int koff = (lane < 16) ? 0 : 2; // K sub-offset for this half-wave

  // Cooperative stage of the shared A tile: 16*K floats, float4 per thread.
  #pragma unroll
  for (int i = threadIdx.x * 4; i < 16 * K; i += 256 * 4) {
    const int r = i / K, c = i % K;     // K is power of two -> shifts
    *(float4*)&As[r * KP + c] =
        *(const float4*)(X + (size_t)(row0 + r) * K + c);
  }
  __syncthreads();

  const float* __restrict__ arow = As + m * KP;
  const float* __restrict__ wrow = W + (size_t)(col0 + m) * K; // B[k][n]=W[n][k]

  v8f acc = {};
#if __has_builtin(__builtin_amdgcn_wmma_f32_16x16x4_f32)
  for (int k = 0; k < K; k += 4) {
    v2f a, b;
    a.x = arow[k + koff];     // A: lane<16 -> K=k,k+1 ; lane>=16 -> K=k+2,k+3
    a.y = arow[k + koff + 1];
    b.x = wrow[k + koff];     // B mirrors the half-wave K split
    b.y = wrow[k + koff + 1];
    acc = __builtin_amdgcn_wmma_f32_16x16x4_f32(
        false, a, false, b, (short)0, acc, false, false);
  }
#else
  for (int k = 0; k < K; ++k) {
    float av = arow[k];
    #pragma unroll
    for (int v = 0; v < 8; ++v) acc[v] += av * W[(size_t)(col0 + v) * K + k];
  }
#endif

  // C/D layout: VGPR v: lanes0-15 -> M=row0+v, lanes16-31 -> M=row0+v+8; N=lane&15
  const int mrow = (lane < 16) ? 0 : 8;
  const int c    = col0 + m;
  const float bv = bias ? bias[c] : 0.0f;
  #pragma unroll
  for (int v = 0; v < 8; ++v) {
    Y[(size_t)(row0 + v + mrow) * DD + c] = acc[v] + bv;
  }
}

// ------------------------- per-node attention logits ------------------------
// e_src[n,h] = dot(xh[n, h*16 : h*16+16], a_src[h]); same for dst.
__global__ __launch_bounds__(256) void att_logits_k(
    const float* __restrict__ xh, const float* __restrict__ a_src,
    const float* __restrict__ a_dst, float* __restrict__ e_src,
    float* __restrict__ e_dst) {
  int t = blockIdx.x * blockDim.x + threadIdx.x;
  if (t >= NN * HH) return;
  int n = t >> 3, h = t & 7;
  const float4* xp = (const float4*)(xh + (size_t)n * DD + h * CC);
  const float4* sp = (const float4*)(a_src + h * CC);
  const float4* dp = (const float4*)(a_dst + h * CC);
  float ss = 0.f, sd = 0.f;
  #pragma unroll
  for (int i = 0; i < 4; ++i) {
    float4 x = xp[i], s = sp[i], d = dp[i];
    ss += x.x * s.x + x.y * s.y + x.z * s.z + x.w * s.w;
    sd += x.x * d.x + x.y * d.y + x.z * d.z + x.w * d.w;
  }
  e_src[t] = ss;
  e_dst[t] = sd;
}

// ------------------------- edge helpers -------------------------------------
__device__ __forceinline__ void edge_nodes(const int* __restrict__ ei, int e,
                                           int& s, int& d) {
  if (e < EE) { s = ei[e]; d = ei[EE + e]; }
  else        { s = e - EE; d = e - EE; }     // appended self-loops
}
__device__ __forceinline__ float leaky(float v) {
  return v > 0.f ? v : v * NEG_SLOPE;
}
// Order-preserving float<->uint map for atomicMax on floats (incl. negatives).
__device__ __forceinline__ unsigned f2ord(float f) {
  unsigned u = __float_as_uint(f);
  return (u & 0x80000000u) ? ~u : (u | 0x80000000u);
}
__device__ __forceinline__ float ord2f(unsigned u) {
  return (u & 0x80000000u) ? __uint_as_float(u & 0x7fffffffu)
                           : __uint_as_float(~u);
}
#define ORD_NEG_INF 0x007FFFFFu   // f2ord(-inf)

// ------------------------- pass 1: segment max over dst ---------------------
__global__ __launch_bounds__(256) void edge_max_k(
    const int* __restrict__ ei, const float* __restrict__ e_src,
    const float* __restrict__ e_dst, unsigned* __restrict__ emax) {
  int t = blockIdx.x * blockDim.x + threadIdx.x;
  if (t >= ETOT * HH) return;
  int e = t >> 3, h = t & 7;
  int s, d; edge_nodes(ei, e, s, d);
  float v = leaky(e_src[s * HH + h] + e_dst[d * HH + h]);
  atomicMax(&emax[d * HH + h], f2ord(v));
}

// ------------------------- pass 2: segment exp-sum over dst -----------------
__global__ __launch_bounds__(256) void edge_den_k(
    const int* __restrict__ ei, const float* __restrict__ e_src,
    const float* __restrict__ e_dst, const unsigned* __restrict__ emax,
    float* __restrict__ denom) {
  int t = blockIdx.x * blockDim.x + threadIdx.x;
  if (t >= ETOT * HH) return;
  int e = t >> 3, h = t & 7;
  int s, d; edge_nodes(ei, e, s, d);
  float v  = leaky(e_src[s * HH + h] + e_dst[d * HH + h]);
  float ex = __expf(v - ord2f(emax[d * HH + h]));
  atomicAdd(&denom[d * HH + h], ex);
}

// ------------------------- pass 3: weighted scatter-add ---------------------
// One edge per 32 lanes; each lane handles 4 channels (float4 gather,
// 4x global_atomic_add_f32 scatter). head = lane>>2.
__global__ __launch_bounds__(256) void edge_aggr_k(
    const int* __restrict__ ei, const float* __restrict__ e_src,
    const float* __restrict__ e_dst, const unsigned* __restrict__ emax,
    const float* __restrict__ denom, const float* __restrict__ xh,
    float* __restrict__ accum) {
  int t = blockIdx.x * blockDim.x + threadIdx.x;
  if (t >= ETOT * 32) return;
  int e = t >> 5, lane = t & 31;
  int h = lane >> 2;
  int s, d; edge_nodes(ei, e, s, d);
  float v     = leaky(e_src[s * HH + h] + e_dst[d * HH + h]);
  float ex    = __expf(v - ord2f(emax[d * HH + h]));
  float alpha = ex / (denom[d * HH + h] + 1e-16f);
  float4 x = *(const float4*)(xh + (size_t)s * DD + lane * 4);
  float* out = accum + (size_t)d * DD + lane * 4;
  atomicAdd(out + 0, x.x * alpha);
  atomicAdd(out + 1, x.y * alpha);
  atomicAdd(out + 2, x.z * alpha);
  atomicAdd(out + 3, x.w * alpha);
}

// ------------------------- residual + bias + LayerNorm ----------------------
// One wave32 per node (4 channels/lane), block = 8 nodes. N % 8 == 0.
__global__ __launch_bounds__(256) void residual_ln_k(
    const float* __restrict__ hin, const float* __restrict__ accum,
    const float* __restrict__ gb, const float* __restrict__ g,
    const float* __restrict__ b, float* __restrict__ hout) {
  int node = blockIdx.x * 8 + (threadIdx.x >> 5);
  int lane = threadIdx.x & 31;
  size_t base = (size_t)node * DD + lane * 4;
  float4 hv = *(const float4*)(hin + base);
  float4 av = *(const float4*)(accum + base);
  float4 bv = *(const float4*)(gb + lane * 4);
  float x0 = hv.x + av.x + bv.x;
  float x1 = hv.y + av.y + bv.y;
  float x2 = hv.z + av.z + bv.z;
  float x3 = hv.w + av.w + bv.w;
  float s = x0 + x1 + x2 + x3;
  #pragma unroll
  for (int o = 16; o > 0; o >>= 1) s += __shfl_xor(s, o, 32);
  float mean = s * (1.f / (float)DD);
  float d0 = x0 - mean, d1 = x1 - mean, d2 = x2 - mean, d3 = x3 - mean;
  float vs = d0 * d0 + d1 * d1 + d2 * d2 + d3 * d3;
  #pragma unroll
  for (int o = 16; o > 0; o >>= 1) vs += __shfl_xor(vs, o, 32);
  float inv = rsqrtf(vs * (1.f / (float)DD) + LN_EPS);
  float4 gv = *(const float4*)(g + lane * 4);
  float4 lb = *(const float4*)(b + lane * 4);
  float4 o4;
  o4.x = d0 * inv * gv.x + lb.x;
  o4.y = d1 * inv * gv.y + lb.y;
  o4.z = d2 * inv * gv.z + lb.z;
  o4.w = d3 * inv * gv.w + lb.w;
  *(float4*)(hout + base) = o4;
}

// ------------------------- mean pool (partial + atomic) ---------------------
__global__ __launch_bounds__(128) void pool_sum_k(const float* __restrict__ h,
                                                  float* __restrict__ pooled) {
  int c = threadIdx.x;  // 0..127
  float s = 0.f;
  for (int n = blockIdx.x; n < NN; n += gridDim.x)
    s += h[(size_t)n * DD + c];
  atomicAdd(&pooled[c], s);
}

// ------------------------- heads: mu / logvar / pooled ----------------------
__global__ __launch_bounds__(128) void head_k(
    const float* __restrict__ pooled_sum, const float* __restrict__ W_mu,
    const float* __restrict__ b_mu, const float* __restrict__ W_lv,
    const float* __restrict__ b_lv, float* __restrict__ out) {
  int d = threadIdx.x;  // 0..127, one block
  __shared__ float pl[DD];
  float pd = pooled_sum[d] * (1.f / (float)NN);
  pl[d] = pd;
  __syncthreads();
  float mu = b_mu[d], lv = b_lv[d];
  for (int k = 0; k < DD; ++k) {
    float p = pl[k];
    mu += p * W_mu[(size_t)d * DD + k];
    lv += p * W_lv[(size_t)d * DD + k];
  }
  out[d]      = mu;                            // mu      [0,128)
  out[DD + d] = lv;                            // logvar  [128,256)
  out[2 * DD + (size_t)NN * DD + d] = pd;      // pooled  (after h)
}

// ------------------------- 32-bit fill --------------------------------------
__global__ __launch_bounds__(256) void fill32_k(unsigned* __restrict__ p,
                                                unsigned v, int n) {
  int t = blockIdx.x * blockDim.x + threadIdx.x;
  if (t < n) p[t] = v;
}

// ---------------------------------------------------------------------------
extern "C" void kernel_launch(void* const* d_in, const int* in_sizes, int n_in,
                              void* d_out, int out_size, void* d_ws, size_t ws_size,
                              hipStream_t stream) {
  const float* x       = (const float*)d_in[0];
  const int*   ei      = (const int*)d_in[1];
  const float* W_in    = (const float*)d_in[2];
  const float* b_in    = (const float*)d_in[3];
  const float* gat_W   = (const float*)d_in[4];   // [L,128,128]
  const float* att_src = (const float*)d_in[5];   // [L,8,16]
  const float* att_dst = (const float*)d_in[6];
  const float* gat_b   = (const float*)d_in[7];   // [L,128]
  const float* ln_g    = (const float*)d_in[8];
  const float* ln_b    = (const float*)d_in[9];
  const float* W_mu    = (const float*)d_in[10];
  const float* b_mu    = (const float*)d_in[11];
  const float* W_lv    = (const float*)d_in[12];
  const float* b_lv    = (const float*)d_in[13];
  float* out = (float*)d_out;

  // Workspace layout (floats): ~166.4 MB total
  float*    h      = (float*)d_ws;            // N*D = 12,800,000
  float*    xh     = h     + (size_t)NN * DD;
  float*    accum  = xh    + (size_t)NN * DD;
  float*    e_src  = accum + (size_t)NN * DD; // N*H = 800,000
  float*    e_dst  = e_src + (size_t)NN * HH;
  unsigned* emax   = (unsigned*)(e_dst + (size_t)NN * HH);
  float*    denom  = (float*)(emax + (size_t)NN * HH);
  float*    pooled = denom + (size_t)NN * HH; // 128

  float* h_final = out + 2 * DD;              // final h lives in d_out

  const int TPB = 256;
  const int nh_blk   = (NN * HH + TPB - 1) / TPB;        // 3125
  const int eh_blk   = (ETOT * HH + TPB - 1) / TPB;      // 53125
  const int ea_blk   = (ETOT * 32 + TPB - 1) / TPB;      // 212500
  const int nd_blk   = ((int)((size_t)NN * DD) + TPB - 1) / TPB; // 50000
  const int ln_blk   = NN / 8;                           // 12500
  const int gemm_blk = NN / 16;                          // 6250

  // h = x @ W_in^T + b_in   (WMMA, K=256, LDS-staged A tile)
  gemm_wmma_k<F_IN><<<gemm_blk, TPB, 0, stream>>>(x, W_in, b_in, h);

  for (int l = 0; l < LL; ++l) {
    const float* Wl  = gat_W   + (size_t)l * DD * DD;
    const float* asl = att_src + (size_t)l * HH * CC;
    const float* adl = att_dst + (size_t)l * HH * CC;
    const float* gbl = gat_b   + (size_t)l * DD;
    const float* gl  = ln_g    + (size_t)l * DD;
    const float* bl  = ln_b    + (size_t)l * DD;

    // xh = h @ W^T   (WMMA, K=128, no bias)
    gemm_wmma_k<DD><<<gemm_blk, TPB, 0, stream>>>(h, Wl, nullptr, xh);

    // per-node attention logits
    att_logits_k<<<nh_blk, TPB, 0, stream>>>(xh, asl, adl, e_src, e_dst);

    // init segment state: emax = ord(-inf), denom = 0, accum = 0
    fill32_k<<<nh_blk, TPB, 0, stream>>>(emax, ORD_NEG_INF, NN * HH);
    fill32_k<<<nh_blk, TPB, 0, stream>>>((unsigned*)denom, 0u, NN * HH);
    fill32_k<<<nd_blk, TPB, 0, stream>>>((unsigned*)accum, 0u, NN * DD);

    // 3-pass dst-softmax + weighted aggregation
    edge_max_k <<<eh_blk, TPB, 0, stream>>>(ei, e_src, e_dst, emax);
    edge_den_k <<<eh_blk, TPB, 0, stream>>>(ei, e_src, e_dst, emax, denom);
    edge_aggr_k<<<ea_blk, TPB, 0, stream>>>(ei, e_src, e_dst, emax, denom, xh, accum);

    // h = LayerNorm(h + accum + gat_b) ; last layer writes straight to d_out
    float* hout = (l == LL - 1) ? h_final : h;
    residual_ln_k<<<ln_blk, TPB, 0, stream>>>(h, accum, gbl, gl, bl, hout);
  }

  // pooled mean + heads
  fill32_k<<<1, TPB, 0, stream>>>((unsigned*)pooled, 0u, DD);
  pool_sum_k<<<512, DD, 0, stream>>>(h_final, pooled);
  head_k<<<1, DD, 0, stream>>>(pooled, W_mu, b_mu, W_lv, b_lv, out);
}